// get_loss_81020263071749
// MI455X (gfx1250) — compile-verified
//
#include <hip/hip_runtime.h>
#include <math.h>

typedef __attribute__((ext_vector_type(2))) float v2f;
typedef __attribute__((ext_vector_type(8))) float v8f;

#define NB        32
#define N_RINGS   10
#define N_ANGLES  36
#define NPTS      360          // N_RINGS * N_ANGLES
#define PTILES    23           // ceil(360/16)
#define NTGT      16384
#define PI_APPROX 3.1415926f

// One wave (32 threads) per (batch, point-tile, target-chunk).
// Computes running min over its chunk of targets for 16 points via
// V_WMMA_F32_16X16X4_F32, writes per-row mins to ws.
__global__ void __launch_bounds__(32)
chamfer_wmma_kernel(const float* __restrict__ pred,
                    const float* __restrict__ target,
                    float* __restrict__ ws,
                    int nsplit, int chunk_len)
{
    const int wid   = blockIdx.x;
    const int chunk = wid % nsplit;
    const int t2    = wid / nsplit;
    const int tile  = t2 % PTILES;
    const int b     = t2 / PTILES;

    const int lane = threadIdx.x;   // 0..31
    const int half = lane >> 4;     // 0 or 1
    const int l    = lane & 15;

    // ---- Build A tile: row M (point) = (-2px, -2py, -2pz, |p|^2) ----
    // 32-bit A 16x4 layout: VGPR0 = {K0 | K2}, VGPR1 = {K1 | K3} across lane halves.
    float a0 = 0.0f, a1 = 0.0f;
    {
        const int p = tile * 16 + l;
        if (p < NPTS) {
            const int j = p / N_ANGLES;
            const int a = p - j * N_ANGLES;
            const float r   = pred[b * N_RINGS + j];
            const float ang = (float)a * (PI_APPROX / 18.0f); // a*10 deg in radians
            float s, c;
            sincosf(ang, &s, &c);
            const float px = -r * c + 0.04f;
            const float py = 0.15f * (float)j - 0.7f;
            const float pz = r * s;
            const float pp = px * px + py * py + pz * pz;
            if (half == 0) { a0 = -2.0f * px; a1 = -2.0f * py; } // K0, K1
            else           { a0 = -2.0f * pz; a1 = pp;         } // K2, K3
        }
    }
    v2f A; A.x = a0; A.y = a1;

    // ---- Stream target chunk: 16 targets per WMMA ----
    const float* tb = target + ((size_t)b * NTGT + (size_t)chunk * chunk_len) * 3;

    float hmin[8];
#pragma unroll
    for (int r = 0; r < 8; ++r) hmin[r] = 3.4e38f;

#pragma unroll 2
    for (int n0 = 0; n0 < chunk_len; n0 += 16) {
        // Lane L and L+16 both load target (n0 + l): xyz
        const float* tp = tb + (size_t)(n0 + l) * 3;
        const float tx = tp[0];
        const float ty = tp[1];
        const float tz = tp[2];
        const float tt = tx * tx + ty * ty + tz * tz;   // |t|^2 for column N=l

        // B 4x16 layout: VGPR0 = {K0 | K2}, VGPR1 = {K1 | K3}
        v2f Bm;
        Bm.x = half ? tz : tx;     // K0 = tx, K2 = tz
        Bm.y = half ? 1.0f : ty;   // K1 = ty, K3 = 1

        v8f C = {};
        C = __builtin_amdgcn_wmma_f32_16x16x4_f32(
                /*neg_a=*/false, A, /*neg_b=*/false, Bm,
                /*c_mod=*/(short)0, C, /*reuse_a=*/false, /*reuse_b=*/false);

        // D[r][lane] = |p|^2 - 2 p.t  for row M=r(+8), column N=l
#pragma unroll
        for (int r = 0; r < 8; ++r) {
            float d2 = C[r] + tt;
            d2 = fmaxf(d2, 0.0f);
            hmin[r] = fminf(hmin[r], d2);
        }
    }

    // ---- Reduce min across the 16 lanes of each half (columns of the tile) ----
#pragma unroll
    for (int r = 0; r < 8; ++r) {
        float v = hmin[r];
        v = fminf(v, __shfl_xor(v, 1, 32));
        v = fminf(v, __shfl_xor(v, 2, 32));
        v = fminf(v, __shfl_xor(v, 4, 32));
        v = fminf(v, __shfl_xor(v, 8, 32));
        // all lanes of each 16-group now hold the row min (row r / row r+8)
        if (l == r) {
            const int M  = half * 8 + r;
            const int gp = tile * 16 + M;
            if (gp < NPTS) {
                ws[((size_t)(b * PTILES + tile) * 16 + M) * nsplit + chunk] = v;
            }
        }
    }
}

// Deterministic final reduction: min over chunks per point, fixed-order sum, mean.
__global__ void __launch_bounds__(256)
chamfer_reduce_kernel(const float* __restrict__ ws,
                      float* __restrict__ out,
                      int nsplit)
{
    __shared__ float sdata[256];
    const int t = threadIdx.x;
    const int total = NB * NPTS;   // 11520

    float s = 0.0f;
    for (int i = t; i < total; i += 256) {
        const int b = i / NPTS;
        const int p = i - b * NPTS;
        const int tile = p >> 4;
        const int M    = p & 15;
        const float* w = ws + ((size_t)(b * PTILES + tile) * 16 + M) * nsplit;
        float m = w[0];
        for (int c = 1; c < nsplit; ++c) m = fminf(m, w[c]);
        s += m;
    }
    sdata[t] = s;
    __syncthreads();
#pragma unroll
    for (int stride = 128; stride > 0; stride >>= 1) {
        if (t < stride) sdata[t] += sdata[t + stride];
        __syncthreads();
    }
    if (t == 0) out[0] = sdata[0] * (1.0f / (float)(NB * NPTS));
}

extern "C" void kernel_launch(void* const* d_in, const int* in_sizes, int n_in,
                              void* d_out, int out_size, void* d_ws, size_t ws_size,
                              hipStream_t stream)
{
    const float* pred   = (const float*)d_in[0];   // (32, 10)
    const float* target = (const float*)d_in[1];   // (32, 16384, 3)
    // d_in[2] (trans_feat) is unused by the reference loss.
    float* ws  = (float*)d_ws;
    float* out = (float*)d_out;

    // Pick target-split for parallelism, bounded by workspace size.
    int nsplit = 8;
    while (nsplit > 1 &&
           (size_t)NB * PTILES * 16 * (size_t)nsplit * sizeof(float) > ws_size) {
        nsplit >>= 1;
    }
    const int chunk_len = NTGT / nsplit;

    const int nblocks = NB * PTILES * nsplit;   // one wave per block
    chamfer_wmma_kernel<<<nblocks, 32, 0, stream>>>(pred, target, ws, nsplit, chunk_len);
    chamfer_reduce_kernel<<<1, 256, 0, stream>>>(ws, out, nsplit);
}